// SuperpixelSAN_50964081935197
// MI455X (gfx1250) — compile-verified
//
#include <hip/hip_runtime.h>
#include <math.h>

// ---------------------------------------------------------------------------
// Types for WMMA f32 16x16x4 (A: 16x4 f32 = 2 VGPRs/lane, B: 4x16 = 2 VGPRs,
// C/D: 16x16 f32 = 8 VGPRs/lane).
// ---------------------------------------------------------------------------
typedef float v2f __attribute__((ext_vector_type(2)));
typedef float v8f __attribute__((ext_vector_type(8)));

// ---------------------------------------------------------------------------
// Dense GEMM with bias: C[M x NC] = A[M x K] * W[K x NC] + bias.
// One wave computes a 16x16 tile via V_WMMA_F32_16X16X4_F32. K and LDC are
// template parameters: the k-loop fully unrolls into back-to-back WMMAs and
// the epilogue becomes 8 unconditional stores with immediate offsets.
// Caller guarantees: C has ceil(M/16)*16 rows and leading dim LDC >= 16*gridDim.y,
// so no store guards are needed (padding columns/rows are junk, never read).
// Loads use clamped in-bounds addresses + select masking -> no divergent
// branches; EXEC stays all-ones at the WMMA (ISA requirement).
// ---------------------------------------------------------------------------
template <int K, int LDC>
__global__ __launch_bounds__(32)
void wmma_gemm_bias(const float* __restrict__ A, int lda,
                    const float* __restrict__ W, int ldw,
                    const float* __restrict__ bias,
                    float* __restrict__ C,
                    int M, int NC)
{
    const int lane = threadIdx.x & 31;
    const int half = lane >> 4;
    const int l16  = lane & 15;
    const int mt = blockIdx.x * 16;
    const int nt = blockIdx.y * 16;
    const int r  = mt + l16;   // A row handled by this lane
    const int n  = nt + l16;   // B/C column handled by this lane
    const bool rok = (r < M);
    const bool nok = (n < NC);
    const int rc  = rok ? r : (M - 1);   // clamped, always in-bounds
    const int ncl = nok ? n : (NC - 1);  // clamped, always in-bounds

    const float* __restrict__ Arow = A + (long)rc * lda;

    v8f c = {0.f, 0.f, 0.f, 0.f, 0.f, 0.f, 0.f, 0.f};

#pragma unroll
    for (int k0 = 0; k0 < K; k0 += 4) {
        const int ka  = k0 + 2 * half;       // {k0,k0+1} or {k0+2,k0+3}
        const int ka0 = (ka     < K) ? ka     : (K - 1);
        const int ka1 = (ka + 1 < K) ? ka + 1 : (K - 1);
        v2f a, b;
        float a0 = Arow[ka0];
        float a1 = Arow[ka1];
        float b0 = W[(long)ka0 * ldw + ncl];
        float b1 = W[(long)ka1 * ldw + ncl];
        a.x = (rok && (ka     < K)) ? a0 : 0.f;
        a.y = (rok && (ka + 1 < K)) ? a1 : 0.f;
        b.x = (nok && (ka     < K)) ? b0 : 0.f;
        b.y = (nok && (ka + 1 < K)) ? b1 : 0.f;
        c = __builtin_amdgcn_wmma_f32_16x16x4_f32(false, a, false, b,
                                                  (short)0, c, false, false);
    }

    float bv = 0.f;
    if (bias != nullptr) {                   // uniform (scalar) branch
        float bl = bias[ncl];
        bv = nok ? bl : 0.f;
    }

    // Unconditional epilogue: rows mt..mt+15 exist (padded alloc), column n < LDC.
    float* __restrict__ Cw = C + (long)(mt + 8 * half) * LDC + n;
#pragma unroll
    for (int i = 0; i < 8; ++i)
        Cw[i * LDC] = c[i] + bv;
}

// ---------------------------------------------------------------------------
// Elementwise / sparse helper kernels
// Feature buffers written by the GEMM (p, hw) have row stride 32; the readout
// buffers (x4, cout) have row stride 16; the atomic accumulator h has stride 30.
// ---------------------------------------------------------------------------
__global__ void k_fill(float* __restrict__ p, float v, long n)
{
    long i = (long)blockIdx.x * blockDim.x + threadIdx.x;
    if (i < n) p[i] = v;
}

// diag[i] += val[e] where row==col
__global__ void k_diag(const int* __restrict__ idx, const float* __restrict__ val,
                       float* __restrict__ diag, int E)
{
    int t = blockIdx.x * blockDim.x + threadIdx.x;
    if (t < E) {
        int r = idx[t], c = idx[E + t];
        if (r == c) atomicAdd(&diag[r], val[t]);
    }
}

// dinv[i] = diag!=0 ? 1/sqrt(|diag|+1e-12) : 0   (in place)
__global__ void k_dinv(float* __restrict__ diag, int N)
{
    int i = blockIdx.x * blockDim.x + threadIdx.x;
    if (i < N) {
        float d = diag[i];
        diag[i] = (d != 0.f) ? (1.0f / sqrtf(fabsf(d) + 1e-12f)) : 0.f;
    }
}

__global__ void k_pval(const int* __restrict__ idx, const float* __restrict__ val,
                       const float* __restrict__ dinv, float* __restrict__ pv, int E)
{
    int t = blockIdx.x * blockDim.x + threadIdx.x;
    if (t < E) pv[t] = val[t] * dinv[idx[t]] * dinv[idx[E + t]];
}

// h[row,:] += pv[e] * x[col,:]  ; one wave per edge, lane = feature (30 of 32)
__global__ void k_spmm(const int* __restrict__ idx, const float* __restrict__ pv,
                       const float* __restrict__ x, float* __restrict__ h, int E)
{
    long g = (long)blockIdx.x * blockDim.x + threadIdx.x;
    int e = (int)(g >> 5), f = (int)(g & 31);
    if (e < E && f < 30) {
        int r = idx[e], c = idx[E + e];
        atomicAdd(&h[(long)r * 30 + f], pv[e] * x[(long)c * 32 + f]);
    }
}

// es/ed = hw . ga_s / ga_d ; also (re)initialize per-node softmax state
__global__ void k_esed(const float* __restrict__ hw,
                       const float* __restrict__ gas, const float* __restrict__ gad,
                       float* __restrict__ es, float* __restrict__ ed,
                       float* __restrict__ m, float* __restrict__ ssum, int N)
{
    int i = blockIdx.x * blockDim.x + threadIdx.x;
    if (i < N) {
        const float* hr = hw + (long)i * 32;
        float s = 0.f, d = 0.f;
#pragma unroll
        for (int j = 0; j < 30; ++j) {
            float v = hr[j];
            s += v * gas[j];
            d += v * gad[j];
        }
        es[i] = s; ed[i] = d;
        m[i] = -INFINITY;
        ssum[i] = 0.f;
    }
}

__device__ inline void atomicMaxF(float* a, float v)
{
    if (v >= 0.f) atomicMax((int*)a, __float_as_int(v));
    else          atomicMin((unsigned int*)a, __float_as_uint(v));
}

// e = leaky_relu(es[row]+ed[col]); store per-edge; segment-max into m[row]
__global__ void k_edge_e(const int* __restrict__ idx,
                         const float* __restrict__ es, const float* __restrict__ ed,
                         float* __restrict__ ebuf, float* __restrict__ m, int E)
{
    int t = blockIdx.x * blockDim.x + threadIdx.x;
    if (t < E) {
        int r = idx[t], c = idx[E + t];
        float v = es[r] + ed[c];
        v = (v > 0.f) ? v : 0.2f * v;
        ebuf[t] = v;
        atomicMaxF(&m[r], v);
    }
}

// ex = exp(e - m[row]); segment-sum into ssum[row]
__global__ void k_edge_exp(const int* __restrict__ idx, float* __restrict__ ebuf,
                           const float* __restrict__ m, float* __restrict__ ssum, int E)
{
    int t = blockIdx.x * blockDim.x + threadIdx.x;
    if (t < E) {
        int r = idx[t];
        float ex = expf(ebuf[t] - m[r]);
        ebuf[t] = ex;
        atomicAdd(&ssum[r], ex);
    }
}

// h[row,:] += (ex / (ssum[row]+1e-16)) * hw[col,:]
__global__ void k_edge_acc(const int* __restrict__ idx, const float* __restrict__ ebuf,
                           const float* __restrict__ ssum, const float* __restrict__ hw,
                           float* __restrict__ h, int E)
{
    long g = (long)blockIdx.x * blockDim.x + threadIdx.x;
    int e = (int)(g >> 5), f = (int)(g & 31);
    if (e < E && f < 30) {
        int r = idx[e], c = idx[E + e];
        float alpha = ebuf[e] / (ssum[r] + 1e-16f);
        atomicAdd(&h[(long)r * 30 + f], alpha * hw[(long)c * 32 + f]);
    }
}

// xcat[i, coloffs+f] = relu(h[i,f] + ngat*gb[f])
__global__ void k_relu_store(const float* __restrict__ h, const float* __restrict__ gb,
                             float ngat, float* __restrict__ xcat,
                             int coloffs, int ldx, int N)
{
    long g = (long)blockIdx.x * blockDim.x + threadIdx.x;
    int i = (int)(g >> 5), f = (int)(g & 31);
    if (i < N && f < 30) {
        float v = h[(long)i * 30 + f] + ngat * gb[f];
        xcat[(long)i * ldx + coloffs + f] = (v > 0.f) ? v : 0.f;
    }
}

// x4 row stride 16 (GEMM-padded)
__global__ void k_pool_acc(const float* __restrict__ x4, const int* __restrict__ batch,
                           float* __restrict__ gsum, float* __restrict__ gcnt, int N)
{
    int i = blockIdx.x * blockDim.x + threadIdx.x;
    if (i < N) {
        int g = batch[i];
        const float* xr = x4 + (long)i * 16;
#pragma unroll
        for (int j = 0; j < 10; ++j) atomicAdd(&gsum[g * 10 + j], xr[j]);
        atomicAdd(&gcnt[g], 1.f);
    }
}

__global__ void k_pool_div(const float* __restrict__ gsum, const float* __restrict__ gcnt,
                           float* __restrict__ G, int coloffs, int Bn)
{
    int t = blockIdx.x * blockDim.x + threadIdx.x;
    if (t < Bn * 10) {
        int g = t / 10, j = t % 10;
        float c = gcnt[g];
        c = (c > 1.f) ? c : 1.f;
        G[g * 30 + coloffs + j] = gsum[t] / c;
    }
}

// in row stride 16 (GEMM-padded), out dense 10
__global__ void k_softmax(const float* __restrict__ in, float* __restrict__ out, int Bn)
{
    int g = blockIdx.x * blockDim.x + threadIdx.x;
    if (g < Bn) {
        const float* r = in + g * 16;
        float mx = r[0];
#pragma unroll
        for (int j = 1; j < 10; ++j) mx = fmaxf(mx, r[j]);
        float e[10], s = 0.f;
#pragma unroll
        for (int j = 0; j < 10; ++j) { e[j] = expf(r[j] - mx); s += e[j]; }
        float* o = out + g * 10;
#pragma unroll
        for (int j = 0; j < 10; ++j) o[j] = e[j] / s;
    }
}

// ---------------------------------------------------------------------------
// Host side
// ---------------------------------------------------------------------------
static inline void* bump(char*& p, size_t bytes)
{
    void* r = (void*)p;
    p += (bytes + 255) & ~(size_t)255;
    return r;
}
static inline int cdiv(long a, long b) { return (int)((a + b - 1) / b); }

extern "C" void kernel_launch(void* const* d_in, const int* in_sizes, int n_in,
                              void* d_out, int out_size, void* d_ws, size_t ws_size,
                              hipStream_t stream)
{
    (void)n_in; (void)out_size; (void)ws_size;

    const float* X[3] = {(const float*)d_in[0], (const float*)d_in[1], (const float*)d_in[2]};
    const int*   L0i  = (const int*)d_in[3];   const float* L0v  = (const float*)d_in[4];
    const int*   L1ui = (const int*)d_in[5];   const float* L1uv = (const float*)d_in[6];
    const int*   L1di = (const int*)d_in[7];   const float* L1dv = (const float*)d_in[8];
    const int*   L2i  = (const int*)d_in[9];   const float* L2v  = (const float*)d_in[10];
    const int*   batch[3] = {(const int*)d_in[11], (const int*)d_in[12], (const int*)d_in[13]};

    const int N[3] = {in_sizes[0] / 3, in_sizes[1] / 3, in_sizes[2] / 3};
    const int E0  = in_sizes[3] / 2, E1u = in_sizes[5] / 2;
    const int E1d = in_sizes[7] / 2, E2  = in_sizes[9] / 2;

    // ---- Resolve the params pytree (two plausible flattening orders) ----
    const float *gW[3][3], *gas[3][3], *gad[3][3], *gb[3][3], *pW[3][3], *pb[3][3];
    const float *finW[3], *finb[3], *combW, *combb;
    const bool sortedOrder = (in_sizes[14] == 300);  // combW (30*10) first if sorted leaves
    if (!sortedOrder) {
        // insertion order: per stack {3x[gW,ga_s,ga_d,gb,pW,pb], finW, finb}, then combW, combb
        int k = 14;
        for (int s = 0; s < 3; ++s) {
            for (int l = 0; l < 3; ++l) {
                gW[s][l]  = (const float*)d_in[k++]; gas[s][l] = (const float*)d_in[k++];
                gad[s][l] = (const float*)d_in[k++]; gb[s][l]  = (const float*)d_in[k++];
                pW[s][l]  = (const float*)d_in[k++]; pb[s][l]  = (const float*)d_in[k++];
            }
            finW[s] = (const float*)d_in[k++]; finb[s] = (const float*)d_in[k++];
        }
        combW = (const float*)d_in[k++]; combb = (const float*)d_in[k++];
    } else {
        // jax sorted-leaves: combW, combb, then per stack {finW, finb, 3x[gW,ga_d,ga_s,gb,pW,pb]}
        int k = 14;
        combW = (const float*)d_in[k++]; combb = (const float*)d_in[k++];
        for (int s = 0; s < 3; ++s) {
            finW[s] = (const float*)d_in[k++]; finb[s] = (const float*)d_in[k++];
            for (int l = 0; l < 3; ++l) {
                gW[s][l]  = (const float*)d_in[k++]; gad[s][l] = (const float*)d_in[k++];
                gas[s][l] = (const float*)d_in[k++]; gb[s][l]  = (const float*)d_in[k++];
                pW[s][l]  = (const float*)d_in[k++]; pb[s][l]  = (const float*)d_in[k++];
            }
        }
    }

    // ---- Workspace layout ----
    char* wp = (char*)d_ws;
    float* Gbuf = (float*)bump(wp, 256 * 30 * 4);   // pooled concat [x0|x1|x2]
    float* gsum = (float*)bump(wp, 256 * 10 * 4);
    float* gcnt = (float*)bump(wp, 256 * 4);
    float* cout = (float*)bump(wp, 256 * 16 * 4);   // comb GEMM out, LDC=16
    char* arena = wp;                                // reused per stack

    struct ES { const int* idx; const float* val; int E; };

    for (int s = 0; s < 3; ++s) {
        ES pe[2]; int npe;
        const int* gidx[2]; int gE[2]; int ngat;
        const int Ns = N[s];
        const float* Xs = X[s];
        if (s == 0)      { pe[0] = {L0i, L0v, E0};   npe = 1;
                           gidx[0] = L0i;  gE[0] = E0;  ngat = 1; }
        else if (s == 1) { pe[0] = {L1ui, L1uv, E1u}; pe[1] = {L1di, L1dv, E1d}; npe = 2;
                           gidx[0] = L1ui; gE[0] = E1u; gidx[1] = L1di; gE[1] = E1d; ngat = 2; }
        else             { pe[0] = {L2i, L2v, E2};   npe = 1;
                           gidx[0] = L2i;  gE[0] = E2;  ngat = 1; }

        long Etot = 0; for (int i = 0; i < npe; ++i) Etot += pe[i].E;
        int Egmax = 0; for (int i = 0; i < ngat; ++i) Egmax = (gE[i] > Egmax) ? gE[i] : Egmax;

        const long Npad = (long)cdiv(Ns, 16) * 16;   // GEMM outputs padded to tile rows

        char* ap = arena;
        float* Pval = (float*)bump(ap, Etot * 4);
        float* diag = (float*)bump(ap, (long)Ns * 4);
        float* xcat = (float*)bump(ap, (long)Ns * 90 * 4);
        float* pbuf = (float*)bump(ap, Npad * 32 * 4);   // GEMM out, LDC=32
        float* hwb  = (float*)bump(ap, Npad * 32 * 4);   // GEMM out, LDC=32
        float* hacc = (float*)bump(ap, (long)Ns * 30 * 4);
        float* es   = (float*)bump(ap, (long)Ns * 4);
        float* ed   = (float*)bump(ap, (long)Ns * 4);
        float* mbuf = (float*)bump(ap, (long)Ns * 4);
        float* ssum = (float*)bump(ap, (long)Ns * 4);
        float* ebuf = (float*)bump(ap, (long)Egmax * 4);
        float* x4   = (float*)bump(ap, Npad * 16 * 4);   // GEMM out, LDC=16

        // ---- symmetric degree normalization of p_val ----
        k_fill<<<cdiv(Ns, 256), 256, 0, stream>>>(diag, 0.f, Ns);
        for (int i = 0; i < npe; ++i)
            k_diag<<<cdiv(pe[i].E, 256), 256, 0, stream>>>(pe[i].idx, pe[i].val, diag, pe[i].E);
        k_dinv<<<cdiv(Ns, 256), 256, 0, stream>>>(diag, Ns);
        float* pv[2]; long eo = 0;
        for (int i = 0; i < npe; ++i) {
            pv[i] = Pval + eo;
            k_pval<<<cdiv(pe[i].E, 256), 256, 0, stream>>>(pe[i].idx, pe[i].val, diag, pv[i], pe[i].E);
            eo += pe[i].E;
        }

        // ---- 3 SAN layers ----
        for (int l = 0; l < 3; ++l) {
            const float* xin = (l == 0) ? Xs : (xcat + (long)(l - 1) * 30);
            const int lda = (l == 0) ? 3 : 90;
            dim3 gdense(cdiv(Ns, 16), 2);  // 30 output cols -> 2 column tiles

            // p = x @ pW + pb ; hw = x @ gW  (K = 3 for layer 0, else 30)
            if (l == 0) {
                wmma_gemm_bias<3, 32><<<gdense, 32, 0, stream>>>(xin, lda, pW[s][l], 30, pb[s][l],
                                                                 pbuf, Ns, 30);
                wmma_gemm_bias<3, 32><<<gdense, 32, 0, stream>>>(xin, lda, gW[s][l], 30, nullptr,
                                                                 hwb, Ns, 30);
            } else {
                wmma_gemm_bias<30, 32><<<gdense, 32, 0, stream>>>(xin, lda, pW[s][l], 30, pb[s][l],
                                                                  pbuf, Ns, 30);
                wmma_gemm_bias<30, 32><<<gdense, 32, 0, stream>>>(xin, lda, gW[s][l], 30, nullptr,
                                                                  hwb, Ns, 30);
            }

            // h = spmm(P, p)
            k_fill<<<cdiv((long)Ns * 30, 256), 256, 0, stream>>>(hacc, 0.f, (long)Ns * 30);
            for (int i = 0; i < npe; ++i)
                k_spmm<<<cdiv((long)pe[i].E * 32, 256), 256, 0, stream>>>(
                    pe[i].idx, pv[i], pbuf, hacc, pe[i].E);

            // GAT passes (both share hw/es/ed: same layer params, per-pass softmax state)
            for (int gi = 0; gi < ngat; ++gi) {
                k_esed<<<cdiv(Ns, 256), 256, 0, stream>>>(hwb, gas[s][l], gad[s][l],
                                                          es, ed, mbuf, ssum, Ns);
                k_edge_e<<<cdiv(gE[gi], 256), 256, 0, stream>>>(gidx[gi], es, ed, ebuf, mbuf, gE[gi]);
                k_edge_exp<<<cdiv(gE[gi], 256), 256, 0, stream>>>(gidx[gi], ebuf, mbuf, ssum, gE[gi]);
                k_edge_acc<<<cdiv((long)gE[gi] * 32, 256), 256, 0, stream>>>(
                    gidx[gi], ebuf, ssum, hwb, hacc, gE[gi]);
            }
            // x_{l+1} = relu(h + ngat*gb), stored into xcat column block l
            k_relu_store<<<cdiv((long)Ns * 32, 256), 256, 0, stream>>>(
                hacc, gb[s][l], (float)ngat, xcat, l * 30, 90, Ns);
        }

        // ---- readout: x4 = [x1|x2|x3] @ finW + finb ; segment-mean over batch ----
        dim3 gfin(cdiv(Ns, 16), 1);
        wmma_gemm_bias<90, 16><<<gfin, 32, 0, stream>>>(xcat, 90, finW[s], 10, finb[s],
                                                        x4, Ns, 10);
        k_fill<<<cdiv(2560, 256), 256, 0, stream>>>(gsum, 0.f, 2560);
        k_fill<<<1, 256, 0, stream>>>(gcnt, 0.f, 256);
        k_pool_acc<<<cdiv(Ns, 256), 256, 0, stream>>>(x4, batch[s], gsum, gcnt, Ns);
        k_pool_div<<<cdiv(2560, 256), 256, 0, stream>>>(gsum, gcnt, Gbuf, s * 10, 256);
    }

    // ---- combine + softmax ----
    dim3 gcomb(16, 1);  // 256 rows, multiple of 16
    wmma_gemm_bias<30, 16><<<gcomb, 32, 0, stream>>>(Gbuf, 30, combW, 10, combb, cout, 256, 10);
    k_softmax<<<1, 256, 0, stream>>>(cout, (float*)d_out, 256);
}